// Model_91010357002433
// MI455X (gfx1250) — compile-verified
//
#include <hip/hip_runtime.h>
#include <hip/hip_bf16.h>

// ---------------------------------------------------------------------------
// Problem constants (match reference)
// ---------------------------------------------------------------------------
#define N_T   100000
#define N_M   20000
#define N_E   600000
#define N_Q   200000
#define F_IN  256
#define DDIM  128

#define K_SLAB   64
#define B_STRIDE 68            // 64 + 4 floats pad: conflict-free half-wave b128 phases

typedef float v2f __attribute__((ext_vector_type(2)));
typedef float v8f __attribute__((ext_vector_type(8)));

// ---------------------------------------------------------------------------
// Degree count: one thread per edge, float atomic add of 1.0
// ---------------------------------------------------------------------------
__global__ void k_deg_count(const int* __restrict__ idx, float* __restrict__ deg, int n) {
    int t = blockIdx.x * blockDim.x + threadIdx.x;
    if (t >= n) return;
    unsafeAtomicAdd(&deg[idx[t]], 1.0f);
}

// deg -> 1/max(deg,1)
__global__ void k_recip(float* __restrict__ deg, int n) {
    int t = blockIdx.x * blockDim.x + threadIdx.x;
    if (t >= n) return;
    deg[t] = 1.0f / fmaxf(deg[t], 1.0f);
}

// ---------------------------------------------------------------------------
// h_m = prof_emb[mentor_node_id] : one float4 per thread
// ---------------------------------------------------------------------------
__global__ void k_gather_rows(const float* __restrict__ src, const int* __restrict__ ids,
                              float* __restrict__ dst, int nrows) {
    int t = blockIdx.x * blockDim.x + threadIdx.x;
    if (t >= nrows * (DDIM / 4)) return;
    int row = t >> 5;          // 32 float4 per row
    int l   = t & 31;
    float4 v = *(const float4*)(src + (size_t)ids[row] * DDIM + l * 4);
    *(float4*)(dst + (size_t)row * DDIM + l * 4) = v;
}

// ---------------------------------------------------------------------------
// Scatter-mean: agg[to[e]] += feat[from[e]] * rdeg[to[e]]
// 32 lanes per edge, float4 per lane, native fp32 global atomics
// ---------------------------------------------------------------------------
__global__ void k_scatter_mean(const float* __restrict__ feat,
                               const int* __restrict__ idx_from,
                               const int* __restrict__ idx_to,
                               const float* __restrict__ rdeg_to,
                               float* __restrict__ agg, int nedges) {
    long long t = (long long)blockIdx.x * blockDim.x + threadIdx.x;
    long long e = t >> 5;
    if (e >= nedges) return;
    int l = (int)(t & 31);
    int s = idx_from[e];
    int d = idx_to[e];
    float r = rdeg_to[d];
    float4 v = *(const float4*)(feat + (size_t)s * DDIM + l * 4);
    float* out = agg + (size_t)d * DDIM + l * 4;
    unsafeAtomicAdd(out + 0, v.x * r);
    unsafeAtomicAdd(out + 1, v.y * r);
    unsafeAtomicAdd(out + 2, v.z * r);
    unsafeAtomicAdd(out + 3, v.w * r);
}

// ---------------------------------------------------------------------------
// One GEMM pass with LDS-staged weights.
//   acc += A[16 rows starting rowA, :K] @ W[128, K]^T  (per-wave 16x128 stripe)
// K processed in 64-wide slabs staged into LDS by the whole 256-thread block.
// Within a slab, K is consumed 8 at a time with effective column order
// {kk,kk+1,kk+4,kk+5},{kk+2,kk+3,kk+6,kk+7}: one b128 A load + one b128 LDS
// B load feed TWO v_wmma_f32_16x16x4_f32 each (fragment layout ISA 7.12.2:
// lane = h*16+r, frag VGPR j holds effective K = 2h+j).
// ---------------------------------------------------------------------------
__device__ __forceinline__ void gemm_pass(const float* __restrict__ A,
                                          const float* __restrict__ W,
                                          int K, int rowA, int r, int h, int tid,
                                          float* __restrict__ Bs, v8f acc[8]) {
    for (int kk0 = 0; kk0 < K; kk0 += K_SLAB) {
        __syncthreads();                       // protect previous slab's readers
        // ---- cooperative stage: 128 x 64 slab of W, padded stride ----
#pragma unroll
        for (int i = 0; i < 8; i++) {
            int idx = tid + i * 256;           // 0..2047 float4 tiles
            int row = idx >> 4;                // 16 float4 per row
            int c   = (idx & 15) << 2;
            float4 v = *(const float4*)(W + (size_t)row * K + kk0 + c);
            *(float4*)(Bs + row * B_STRIDE + c) = v;
        }
        __syncthreads();

        const float* Arow = A + (size_t)(rowA + r) * K + kk0 + 4 * h;
        for (int ks = 0; ks < K_SLAB; ks += 8) {
            float4 a4 = *(const float4*)(Arow + ks);
            v2f a1; a1[0] = a4.x; a1[1] = a4.y;
            v2f a2; a2[0] = a4.z; a2[1] = a4.w;
#pragma unroll
            for (int nt = 0; nt < 8; nt++) {
                float4 b4 = *(const float4*)(Bs + (nt * 16 + r) * B_STRIDE + ks + 4 * h);
                v2f b1; b1[0] = b4.x; b1[1] = b4.y;
                v2f b2; b2[0] = b4.z; b2[1] = b4.w;
                acc[nt] = __builtin_amdgcn_wmma_f32_16x16x4_f32(
                    false, a1, false, b1, (short)0, acc[nt], false, false);
                acc[nt] = __builtin_amdgcn_wmma_f32_16x16x4_f32(
                    false, a2, false, b2, (short)0, acc[nt], false, false);
            }
        }
    }
}

// out[nrows,128] = A1 @ W1^T (+ A2 @ W2^T) + bias, optional ReLU
__global__ void __launch_bounds__(256)
k_gemm_wmma(const float* __restrict__ A1, const float* __restrict__ W1, int K1,
            const float* __restrict__ A2, const float* __restrict__ W2, int K2,
            const float* __restrict__ bias, float* __restrict__ out,
            int nrows, int relu) {
    __shared__ float Bs[DDIM * B_STRIDE];      // 34.8 KB

    const int tid  = threadIdx.x;
    const int lane = tid & 31;
    const int wave = tid >> 5;
    const int row0 = (blockIdx.x * 8 + wave) * 16;
    const bool valid = (row0 < nrows);
    const int rowA = valid ? row0 : 0;         // invalid waves compute on row 0,
                                               // keep barriers + EXEC all-ones
    const int r = lane & 15;
    const int h = lane >> 4;

    v8f acc[8] = {};

    gemm_pass(A1, W1, K1, rowA, r, h, tid, Bs, acc);
    if (A2) gemm_pass(A2, W2, K2, rowA, r, h, tid, Bs, acc);   // block-uniform branch

    if (!valid) return;
#pragma unroll
    for (int nt = 0; nt < 8; nt++) {
        int col  = nt * 16 + r;
        float bb = bias[col];
#pragma unroll
        for (int i = 0; i < 8; i++) {
            // C/D layout: VGPR i, lanes 0-15 -> M=i; lanes 16-31 -> M=8+i; N=r
            float v = acc[nt][i] + bb;
            if (relu) v = fmaxf(v, 0.0f);
            out[(size_t)(row0 + h * 8 + i) * DDIM + col] = v;
        }
    }
}

// ---------------------------------------------------------------------------
// Classifier: out[q] = dot(ht[src], w[0:128]) + dot(hm[dst], w[128:256]) + b
// One wave per query, float4 per lane, wave32 xor-shuffle reduction.
// ---------------------------------------------------------------------------
__global__ void k_classify(const float* __restrict__ ht, const float* __restrict__ hm,
                           const int* __restrict__ qs, const int* __restrict__ qd,
                           const float* __restrict__ w, const float* __restrict__ cb,
                           float* __restrict__ out, int nq) {
    long long t = (long long)blockIdx.x * blockDim.x + threadIdx.x;
    long long q = t >> 5;
    if (q >= nq) return;
    int l = (int)(t & 31);
    float4 a  = *(const float4*)(ht + (size_t)qs[q] * DDIM + l * 4);
    float4 b  = *(const float4*)(hm + (size_t)qd[q] * DDIM + l * 4);
    float4 wa = *(const float4*)(w + l * 4);
    float4 wb = *(const float4*)(w + DDIM + l * 4);
    float s = a.x * wa.x + a.y * wa.y + a.z * wa.z + a.w * wa.w
            + b.x * wb.x + b.y * wb.y + b.z * wb.z + b.w * wb.w;
#pragma unroll
    for (int off = 16; off > 0; off >>= 1) s += __shfl_xor(s, off, 32);
    if (l == 0) out[q] = s + cb[0];
}

// ---------------------------------------------------------------------------
// Host orchestration
// ---------------------------------------------------------------------------
extern "C" void kernel_launch(void* const* d_in, const int* in_sizes, int n_in,
                              void* d_out, int out_size, void* d_ws, size_t ws_size,
                              hipStream_t stream) {
    const float* x_thesis     = (const float*)d_in[0];
    const int*   mentor_id    = (const int*)  d_in[1];
    const int*   edge_src     = (const int*)  d_in[2];
    const int*   edge_dst     = (const int*)  d_in[3];
    const int*   eli_src      = (const int*)  d_in[4];
    const int*   eli_dst      = (const int*)  d_in[5];
    const float* thesis_lin_w = (const float*)d_in[6];
    const float* thesis_lin_b = (const float*)d_in[7];
    const float* prof_emb     = (const float*)d_in[8];
    const float* W_nbr_sup    = (const float*)d_in[9];
    const float* W_root_sup   = (const float*)d_in[10];
    const float* b_sup        = (const float*)d_in[11];
    const float* W_nbr_rev    = (const float*)d_in[12];
    const float* W_root_rev   = (const float*)d_in[13];
    const float* b_rev        = (const float*)d_in[14];
    const float* cls_w        = (const float*)d_in[15];
    const float* cls_b        = (const float*)d_in[16];
    float*       out          = (float*)d_out;

    // ---- workspace carve-up (~185 MB) ----
    float* p    = (float*)d_ws;
    float* ht0  = p; p += (size_t)N_T * DDIM;
    float* ht1  = p; p += (size_t)N_T * DDIM;
    float* aggt = p; p += (size_t)N_T * DDIM;
    float* hm0  = p; p += (size_t)N_M * DDIM;
    float* hm1  = p; p += (size_t)N_M * DDIM;
    float* aggm = p; p += (size_t)N_M * DDIM;
    float* rdt  = p; p += N_T;
    float* rdm  = p; p += N_M;

    const int BLK = 256;
    const int ROWS_PER_BLK = 8 * 16;            // 8 waves x 16 rows

    // ---- degrees (once) ----
    hipMemsetAsync(rdt, 0, (size_t)N_T * sizeof(float), stream);
    hipMemsetAsync(rdm, 0, (size_t)N_M * sizeof(float), stream);
    k_deg_count<<<(N_E + BLK - 1) / BLK, BLK, 0, stream>>>(edge_dst, rdm, N_E);
    k_deg_count<<<(N_E + BLK - 1) / BLK, BLK, 0, stream>>>(edge_src, rdt, N_E);
    k_recip<<<(N_M + BLK - 1) / BLK, BLK, 0, stream>>>(rdm, N_M);
    k_recip<<<(N_T + BLK - 1) / BLK, BLK, 0, stream>>>(rdt, N_T);

    // ---- node encoders ----
    k_gemm_wmma<<<(N_T + ROWS_PER_BLK - 1) / ROWS_PER_BLK, BLK, 0, stream>>>(
        x_thesis, thesis_lin_w, F_IN, nullptr, nullptr, 0,
        thesis_lin_b, ht0, N_T, 0);
    {
        int n = N_M * (DDIM / 4);
        k_gather_rows<<<(n + BLK - 1) / BLK, BLK, 0, stream>>>(prof_emb, mentor_id, hm0, N_M);
    }

    const long long scat_threads = (long long)N_E * 32;
    const int scat_blocks = (int)((scat_threads + BLK - 1) / BLK);

    // ---- layer 0 (ReLU) ----
    hipMemsetAsync(aggm, 0, (size_t)N_M * DDIM * sizeof(float), stream);
    k_scatter_mean<<<scat_blocks, BLK, 0, stream>>>(ht0, edge_src, edge_dst, rdm, aggm, N_E);
    hipMemsetAsync(aggt, 0, (size_t)N_T * DDIM * sizeof(float), stream);
    k_scatter_mean<<<scat_blocks, BLK, 0, stream>>>(hm0, edge_dst, edge_src, rdt, aggt, N_E);
    k_gemm_wmma<<<(N_M + ROWS_PER_BLK - 1) / ROWS_PER_BLK, BLK, 0, stream>>>(
        aggm, W_nbr_sup, DDIM, hm0, W_root_sup, DDIM, b_sup, hm1, N_M, 1);
    k_gemm_wmma<<<(N_T + ROWS_PER_BLK - 1) / ROWS_PER_BLK, BLK, 0, stream>>>(
        aggt, W_nbr_rev, DDIM, ht0, W_root_rev, DDIM, b_rev, ht1, N_T, 1);

    // ---- layer 1 (no ReLU) ----
    const size_t wOff = (size_t)DDIM * DDIM;    // layer-1 weight offset
    hipMemsetAsync(aggm, 0, (size_t)N_M * DDIM * sizeof(float), stream);
    k_scatter_mean<<<scat_blocks, BLK, 0, stream>>>(ht1, edge_src, edge_dst, rdm, aggm, N_E);
    hipMemsetAsync(aggt, 0, (size_t)N_T * DDIM * sizeof(float), stream);
    k_scatter_mean<<<scat_blocks, BLK, 0, stream>>>(hm1, edge_dst, edge_src, rdt, aggt, N_E);
    k_gemm_wmma<<<(N_M + ROWS_PER_BLK - 1) / ROWS_PER_BLK, BLK, 0, stream>>>(
        aggm, W_nbr_sup + wOff, DDIM, hm1, W_root_sup + wOff, DDIM, b_sup + DDIM, hm0, N_M, 0);
    k_gemm_wmma<<<(N_T + ROWS_PER_BLK - 1) / ROWS_PER_BLK, BLK, 0, stream>>>(
        aggt, W_nbr_rev + wOff, DDIM, ht1, W_root_rev + wOff, DDIM, b_rev + DDIM, ht0, N_T, 0);

    // ---- classifier ----
    {
        long long t = (long long)N_Q * 32;
        k_classify<<<(int)((t + BLK - 1) / BLK), BLK, 0, stream>>>(
            ht0, hm0, eli_src, eli_dst, cls_w, cls_b, out, N_Q);
    }
}